// UniformRegularizationLoss_33500744908969
// MI455X (gfx1250) — compile-verified
//
#include <hip/hip_runtime.h>

// ---------------------------------------------------------------------------
// Sinkhorn + uniform-regularization loss for MI455X (gfx1250, wave32).
//
// B = 8192, 50 iterations -> ~26 GB HBM traffic -> ~1.1 ms floor at 23.3 TB/s.
// exp() (trans-rate VALU) throughput is comparable to the memory time, so the
// row reduction runs on the matrix pipe (V_WMMA_F32_16X16X4_F32, ones
// B-matrix) and data is staged coalesced into LDS with the gfx1250 async
// global->LDS path (ASYNCcnt), double-buffered against the exp+WMMA compute.
// ---------------------------------------------------------------------------

#define B_DIM 8192
#define BETA 0.3f
#define NUM_ITERS 50
// log(m+n) = log(16384) = 14*ln2
#define LOG_NORM 9.70406052784f
#define LOG_MU (-LOG_NORM)
#define LOG_NU (-LOG_NORM)

typedef float v2f __attribute__((ext_vector_type(2)));
typedef float v8f __attribute__((ext_vector_type(8)));
typedef int   v4i __attribute__((ext_vector_type(4)));

#define TILE_K     64
#define LDS_STRIDE 68   // 64 + 4 pad -> strided 16-row reads hit distinct banks

#if __has_builtin(__builtin_amdgcn_global_load_async_to_lds_b128) && \
    __has_builtin(__builtin_amdgcn_s_wait_asynccnt)
#define HAVE_ASYNC_LDS 1
#else
#define HAVE_ASYNC_LDS 0
#endif

#if HAVE_ASYNC_LDS
// 16B global -> LDS async copy (per-lane). Builtin signature (from hipcc
// diagnostics): param0 = v4i* in the global (AS1) address space, param1 =
// LDS destination, then immediate offset + cpol. Pointer casts go via
// integers: p1 keeps the full 64-bit VA; p3 takes the low 32 bits (LDS byte
// offset), matching addrspacecast generic->local on AMDGPU.
__device__ __forceinline__ void async_copy16(const float* g, float* l) {
    __builtin_amdgcn_global_load_async_to_lds_b128(
        (__attribute__((address_space(1))) v4i*)(unsigned long long)g,
        (__attribute__((address_space(3))) v4i*)(unsigned)(unsigned long long)l,
        0, 0);
}
#endif

// Workspace layout (floats):
//   [0 , B)   u
//   [B , 2B)  v
//   [2B, 3B)  rowsum accumulator (reset by finalize)
//   [3B, 4B)  colsum accumulator (reset by finalize)
//   [4B, 5B)  row_lse = logsumexp(S, axis=1) (saved on iteration 0)

__global__ __launch_bounds__(256) void sinkhorn_init(float* ws, float* out) {
    int i = blockIdx.x * 256 + threadIdx.x;
    if (i < 4 * B_DIM) ws[i] = 0.0f;   // u, v, rowsum, colsum
    if (i == 0) out[0] = 0.0f;
}

// Row pass: rowsum[i] += sum_{k in chunk} exp(S[i][k] + v[k]).
// One wave per 16-row strip; gridDim.y splits K into 4 chunks of 2048.
// Pipeline per 16x64 tile: async coalesced global->LDS (8x b128/lane, 512B
// per wave per instr), s_wait_asynccnt, then ds_load_b64 in the WMMA A
// layout (lane l<16: row l, K=k,k+1; lane l>=16: row l-16, K=k+2,k+3),
// exp on VALU, accumulate via V_WMMA_F32_16X16X4_F32 with B = ones(4x16).
__global__ __launch_bounds__(32)
void sinkhorn_row_partial(const float* __restrict__ S,
                          const float* __restrict__ v,
                          float* __restrict__ rowsum) {
    const int row0 = blockIdx.x * 16;
    const int k0   = blockIdx.y * (B_DIM / 4);
    const int lane = threadIdx.x;
    const int mrow = lane & 15;
    const int koff = (lane >> 4) << 1;   // 0 for lanes 0-15, 2 for lanes 16-31

    v8f c = {};                          // f32 accumulator (8 VGPRs)
    v2f b; b.x = 1.0f; b.y = 1.0f;       // ones B-matrix (4x16)

#if HAVE_ASYNC_LDS
    __shared__ float tile[2][16][LDS_STRIDE];
    const int rpar = lane >> 4;          // staging row parity
    const int kl   = lane & 15;          // staging k-lane (16B each)

    auto stage = [&](int buf, int kbase) {
        #pragma unroll
        for (int i = 0; i < 8; ++i) {    // instr i covers rows 2i, 2i+1
            const int rt = 2 * i + rpar;
            const float* g = S + (size_t)(row0 + rt) * B_DIM + kbase + kl * 4;
            async_copy16(g, &tile[buf][rt][kl * 4]);
        }
    };

    const int ntiles = (B_DIM / 4) / TILE_K;   // 32
    stage(0, k0);
    for (int t = 0; t < ntiles; ++t) {
        if (t + 1 < ntiles) {
            stage((t + 1) & 1, k0 + (t + 1) * TILE_K);
            __builtin_amdgcn_s_wait_asynccnt(8);   // tile t complete
        } else {
            __builtin_amdgcn_s_wait_asynccnt(0);
        }
        const int buf = t & 1;
        const float* __restrict__ vt = v + k0 + t * TILE_K;
        #pragma unroll 4
        for (int kk = 0; kk < TILE_K; kk += 4) {
            const float2 sv = *(const float2*)&tile[buf][mrow][kk + koff];
            const float2 vv = *(const float2*)(vt + kk + koff);
            v2f a;
            a.x = __expf(sv.x + vv.x);
            a.y = __expf(sv.y + vv.y);
            c = __builtin_amdgcn_wmma_f32_16x16x4_f32(
                    false, a, false, b, (short)0, c, false, false);
        }
    }
#else
    // Fallback: feed WMMA straight from global (cache-served strided reads).
    const int row = row0 + mrow;
    const float* __restrict__ srow = S + (size_t)row * B_DIM;
    const int kend = k0 + (B_DIM / 4);
    #pragma unroll 4
    for (int k = k0; k < kend; k += 4) {
        const float2 sv = *(const float2*)(srow + k + koff);
        const float2 vv = *(const float2*)(v + k + koff);
        v2f a;
        a.x = __expf(sv.x + vv.x);
        a.y = __expf(sv.y + vv.y);
        c = __builtin_amdgcn_wmma_f32_16x16x4_f32(
                false, a, false, b, (short)0, c, false, false);
    }
#endif

    // C/D layout: VGPR r, lanes 0-15 -> M=r, lanes 16-31 -> M=r+8 (any N).
    if ((lane & 15) == 0) {
        const int rbase = row0 + ((lane >> 4) << 3);
        #pragma unroll
        for (int r = 0; r < 8; ++r)
            atomicAdd(&rowsum[rbase + r], c[r]);
    }
}

// Column pass: colsum[j] += sum_{i in chunk} exp(S[i][j] + u[i]).
// 4 columns per thread (b128 loads, fully coalesced); gridDim.y splits rows.
__global__ __launch_bounds__(256)
void sinkhorn_col_partial(const float* __restrict__ S,
                          const float* __restrict__ u,
                          float* __restrict__ colsum) {
    const int col = (blockIdx.x * 256 + threadIdx.x) * 4;
    const int r0  = blockIdx.y * (B_DIM / 32);
    const int r1  = r0 + (B_DIM / 32);
    float4 acc = make_float4(0.f, 0.f, 0.f, 0.f);
    for (int i = r0; i < r1; ++i) {
        const float* p = S + (size_t)i * B_DIM + col;
        __builtin_prefetch(p + 8 * B_DIM, 0, 0);       // global_prefetch_b8
        const float4 s = *(const float4*)p;
        const float ui = u[i];
        acc.x += __expf(s.x + ui);
        acc.y += __expf(s.y + ui);
        acc.z += __expf(s.z + ui);
        acc.w += __expf(s.w + ui);
    }
    atomicAdd(&colsum[col + 0], acc.x);
    atomicAdd(&colsum[col + 1], acc.y);
    atomicAdd(&colsum[col + 2], acc.z);
    atomicAdd(&colsum[col + 3], acc.w);
}

// dst[i] = log_const - log(acc[i]); optionally save lse; reset accumulator.
__global__ __launch_bounds__(256)
void sinkhorn_finalize(float* __restrict__ dst,
                       float* __restrict__ acc,
                       float* __restrict__ lse_out,
                       float log_const, int save_lse) {
    int i = blockIdx.x * 256 + threadIdx.x;
    if (i < B_DIM) {
        float lse = __logf(acc[i]);
        dst[i] = log_const - lse;
        if (save_lse) lse_out[i] = lse;
        acc[i] = 0.0f;                 // ready for next iteration (graph-safe)
    }
}

// loss = -(1/B) * sum_i sum_j [beta*exp(S+u_i+v_j-logN) + (1-beta)*delta_ij]
//                            * (S_ij - row_lse_i)
__global__ __launch_bounds__(256)
void sinkhorn_loss(const float* __restrict__ S,
                   const float* __restrict__ u,
                   const float* __restrict__ v,
                   const float* __restrict__ row_lse,
                   float* __restrict__ out) {
    const int row = blockIdx.x;
    const float lse = row_lse[row];
    const float ui  = u[row];
    const float* __restrict__ srow = S + (size_t)row * B_DIM;

    float acc = 0.0f;
    for (int j = threadIdx.x * 4; j < B_DIM; j += 1024) {
        const float4 s  = *(const float4*)(srow + j);
        const float4 vv = *(const float4*)(v + j);
        float lp, t;
        lp = s.x - lse;
        t  = BETA * __expf(s.x + ui + vv.x - LOG_NORM) + ((j + 0 == row) ? (1.0f - BETA) : 0.0f);
        acc += t * lp;
        lp = s.y - lse;
        t  = BETA * __expf(s.y + ui + vv.y - LOG_NORM) + ((j + 1 == row) ? (1.0f - BETA) : 0.0f);
        acc += t * lp;
        lp = s.z - lse;
        t  = BETA * __expf(s.z + ui + vv.z - LOG_NORM) + ((j + 2 == row) ? (1.0f - BETA) : 0.0f);
        acc += t * lp;
        lp = s.w - lse;
        t  = BETA * __expf(s.w + ui + vv.w - LOG_NORM) + ((j + 3 == row) ? (1.0f - BETA) : 0.0f);
        acc += t * lp;
    }
    // wave32 reduce, then cross-wave via LDS
    for (int off = 16; off > 0; off >>= 1)
        acc += __shfl_down(acc, off, 32);
    __shared__ float red[8];
    if ((threadIdx.x & 31) == 0) red[threadIdx.x >> 5] = acc;
    __syncthreads();
    if (threadIdx.x == 0) {
        float t = 0.0f;
        #pragma unroll
        for (int w = 0; w < 8; ++w) t += red[w];
        atomicAdd(out, -t * (1.0f / (float)B_DIM));
    }
}

extern "C" void kernel_launch(void* const* d_in, const int* in_sizes, int n_in,
                              void* d_out, int out_size, void* d_ws, size_t ws_size,
                              hipStream_t stream) {
    const float* S = (const float*)d_in[0];
    float* ws      = (float*)d_ws;
    float* u       = ws;
    float* v       = ws + B_DIM;
    float* rowsum  = ws + 2 * B_DIM;
    float* colsum  = ws + 3 * B_DIM;
    float* row_lse = ws + 4 * B_DIM;
    float* out     = (float*)d_out;

    sinkhorn_init<<<(4 * B_DIM + 255) / 256, 256, 0, stream>>>(ws, out);

    for (int it = 0; it < NUM_ITERS; ++it) {
        // u = log_mu - logsumexp(S + v[None,:], axis=1)
        sinkhorn_row_partial<<<dim3(B_DIM / 16, 4), 32, 0, stream>>>(S, v, rowsum);
        sinkhorn_finalize<<<B_DIM / 256, 256, 0, stream>>>(
            u, rowsum, row_lse, LOG_MU, it == 0 ? 1 : 0);
        // v = log_nu - logsumexp(S + u[:,None], axis=0)
        sinkhorn_col_partial<<<dim3(B_DIM / 1024, 32), 256, 0, stream>>>(S, u, colsum);
        sinkhorn_finalize<<<B_DIM / 256, 256, 0, stream>>>(
            v, colsum, nullptr, LOG_NU, 0);
    }

    sinkhorn_loss<<<B_DIM, 256, 0, stream>>>(S, u, v, row_lse, out);
}